// LocalCosineSimilarity_38706245272209
// MI455X (gfx1250) — compile-verified
//
#include <hip/hip_runtime.h>

// Problem constants (from the reference)
#define B_    4
#define N1_   1024
#define DIM_  1024
#define F_    4
#define NN_   1025            // N = N1 + 1
#define NTOT_ (F_ * NN_)      // 4100
#define NEGV  (-1.0e9f)

typedef float v2f __attribute__((ext_vector_type(2)));
typedef float v8f __attribute__((ext_vector_type(8)));

// ---------------------------------------------------------------------------
// Kernel 1: bandwidth-bound fill of the sim output with -1e9 (float4 stores).
// 16,793,600 floats = 4,198,400 float4 — divisible, base 16B-aligned.
// ---------------------------------------------------------------------------
__global__ void lcs_fill_kernel(float4* __restrict__ out, long n4) {
    long idx    = (long)blockIdx.x * blockDim.x + threadIdx.x;
    long stride = (long)gridDim.x * blockDim.x;
    const float4 v = make_float4(NEGV, NEGV, NEGV, NEGV);
    for (long i = idx; i < n4; i += stride)
        out[i] = v;
}

// ---------------------------------------------------------------------------
// Kernel 2: surviving-diagonal cosine similarity via V_WMMA_F32_16X16X4_F32.
//
// One wave handles (b, f, 16-row tile). Per K-step of 4, each lane loads a
// float2 from its lhs row and its (shifted) rhs row; per ISA layout these
// register pairs are simultaneously valid A-fragments and B-fragments, so:
//   c_lr += A_lhs * B_rhs   -> C[n][n] = dot(lhs_i, rhs_{fN+i+1})
//   c_ll += A_lhs * B_lhs   -> C[n][n] = ||lhs_i||^2
//   c_rr += A_rhs * B_rhs   -> C[n][n] = ||rhs_j||^2
// sim = dot^2 / (ll * rr);  lhs_norm = sqrt(ll).
// ---------------------------------------------------------------------------
__global__ __launch_bounds__(128) void lcs_diag_kernel(
        const float* __restrict__ lhs,
        const float* __restrict__ rhs,
        float* __restrict__ out) {
    const int t = blockIdx.x;            // row tile 0..63 (16 rows each)
    const int b = blockIdx.y;            // batch 0..3
    const int f = threadIdx.x >> 5;      // wave id -> frame 0..3
    const int L = threadIdx.x & 31;      // lane
    const int g = L >> 4;                // lane group (K-half)
    const int m = L & 15;                // row / column index within tile

    const int i_base = t * 16;
    const float* lrow = lhs + ((size_t)(b * N1_   +            i_base + m    ) * DIM_);
    const float* rrow = rhs + ((size_t)(b * NTOT_ + f * NN_ +  i_base + m + 1) * DIM_);

    v8f c_lr = {};   // lhs . rhs
    v8f c_ll = {};   // lhs . lhs
    v8f c_rr = {};   // rhs . rhs

    #pragma unroll 4
    for (int k = 0; k < DIM_; k += 4) {
        const int off = k + 2 * g;       // A 16x4 f32 layout: VGPR0/1 = K(2g), K(2g+1)
        v2f a  = *reinterpret_cast<const v2f*>(lrow + off);
        v2f bb = *reinterpret_cast<const v2f*>(rrow + off);
        c_lr = __builtin_amdgcn_wmma_f32_16x16x4_f32(false, a,  false, bb, (short)0, c_lr, false, false);
        c_ll = __builtin_amdgcn_wmma_f32_16x16x4_f32(false, a,  false, a,  (short)0, c_ll, false, false);
        c_rr = __builtin_amdgcn_wmma_f32_16x16x4_f32(false, bb, false, bb, (short)0, c_rr, false, false);
    }

    // Diagonal extraction: lane L holds D[M=r+8g][n=L&15]; diag needs r = n-8g.
    const int n = m;
    float dot = 0.f, ll = 0.f, rr = 0.f;
    #pragma unroll
    for (int r = 0; r < 8; ++r) {
        const bool sel = (n == r + 8 * g);
        dot = sel ? c_lr[r] : dot;
        ll  = sel ? c_ll[r] : ll;
        rr  = sel ? c_rr[r] : rr;
    }
    const bool valid = (g == 0) ? (n < 8) : (n >= 8);

    if (valid) {
        const int i = i_base + n;                               // 0..1023
        const float sim = (dot * dot) / (ll * rr);              // cos^2
        out[(size_t)(b * N1_ + i) * NTOT_ + f * NN_ + (i + 1)] = sim;
        if (f == 0) {
            // second output: lhs_norm (B, N1, 1), concatenated after sim
            out[(size_t)B_ * N1_ * NTOT_ + b * N1_ + i] = __builtin_sqrtf(ll);
        }
    }
}

// ---------------------------------------------------------------------------
extern "C" void kernel_launch(void* const* d_in, const int* in_sizes, int n_in,
                              void* d_out, int out_size, void* d_ws, size_t ws_size,
                              hipStream_t stream) {
    const float* lhs = (const float*)d_in[0];   // (B, N1, DIM) f32
    const float* rhs = (const float*)d_in[1];   // (B, NTOT, DIM) f32
    float* out = (float*)d_out;                 // sim (B,N1,NTOT) ++ lhs_norm (B,N1)

    const long n4 = (long)B_ * N1_ * NTOT_ / 4; // 4,198,400 float4
    lcs_fill_kernel<<<2048, 256, 0, stream>>>((float4*)out, n4);

    dim3 grid(N1_ / 16, B_);                    // (64 tiles, 4 batches)
    lcs_diag_kernel<<<grid, 128, 0, stream>>>(lhs, rhs, out);
}